// CAMModule_87316685127757
// MI455X (gfx1250) — compile-verified
//
#include <hip/hip_runtime.h>

typedef __attribute__((ext_vector_type(2))) float v2f;
typedef __attribute__((ext_vector_type(4))) float v4f;
typedef __attribute__((ext_vector_type(8))) float v8f;
typedef __attribute__((ext_vector_type(4))) int   v4i;

constexpr int Cc = 512;    // channels
constexpr int Nn = 4096;   // h*w
constexpr int KB = 32;     // K-block staged through LDS
constexpr int LP = KB + 4; // padded LDS row stride (36 dwords = 144 B, 16B-aligned, conflict-free)

#if defined(__AMDGCN__) && __has_builtin(__builtin_amdgcn_global_load_async_to_lds_b128)
#define CAM_ASYNC_LDS 1
#else
#define CAM_ASYNC_LDS 0
#endif

// Stage 16 bytes global -> LDS. Async (ASYNCcnt, no VGPR round-trip) when available.
__device__ __forceinline__ void stage_b128(const float* gsrc, float* ldst) {
#if CAM_ASYNC_LDS
    __builtin_amdgcn_global_load_async_to_lds_b128(
        (__attribute__((address_space(1))) v4i*)(const_cast<float*>(gsrc)),
        (__attribute__((address_space(3))) v4i*)ldst,
        0, 0);
#else
    *(v4f*)ldst = *(const v4f*)gsrc;
#endif
}

__device__ __forceinline__ void stage_wait() {
#if CAM_ASYNC_LDS
#if __has_builtin(__builtin_amdgcn_s_wait_asynccnt)
    __builtin_amdgcn_s_wait_asynccnt(0);
#else
    asm volatile("s_wait_asynccnt 0" ::: "memory");
#endif
#endif
}

// ---------------------------------------------------------------------------
// Kernel 1: energy[b] = Q[b] * Q[b]^T   (Q = x viewed as [C][N] per batch)
// 64x64 output tile per WG, 8 waves, 2x 16x16 f32 accumulators per wave.
// Double-buffered LDS: async copy of K-block i+1 overlaps WMMA on block i.
// ---------------------------------------------------------------------------
__global__ __launch_bounds__(256)
void cam_energy_kernel(const float* __restrict__ x, float* __restrict__ energy) {
    const int b     = blockIdx.z;
    const int tileM = blockIdx.y * 64;
    const int tileN = blockIdx.x * 64;
    const float* Q = x + (size_t)b * Cc * Nn;
    float*       E = energy + (size_t)b * Cc * Cc;

    __shared__ __align__(16) float As[2][64][LP];
    __shared__ __align__(16) float Bs[2][64][LP];

    const int tid   = threadIdx.x;
    const int lane  = tid & 31;
    const int wave  = tid >> 5;
    const int waveM = (wave & 3) * 16;   // 0..48
    const int waveN = (wave >> 2) * 32;  // 0 or 32

    const int frow = tid >> 3;           // 0..31 (LDS fill)
    const int fcol = (tid & 7) * 4;      // 0..28

    const int mA = waveM + (lane & 15);
    const int nB = waveN + (lane & 15);
    const int kh = (lane >> 4) * 2;      // 0 or 2: this lane's K pair

    v8f acc0 = {}; v8f acc1 = {};

    const float* ga0 = Q + (size_t)(tileM + frow)      * Nn + fcol;
    const float* ga1 = Q + (size_t)(tileM + frow + 32) * Nn + fcol;
    const float* gb0 = Q + (size_t)(tileN + frow)      * Nn + fcol;
    const float* gb1 = Q + (size_t)(tileN + frow + 32) * Nn + fcol;

    // prologue: stage K-block 0 into buffer 0
    stage_b128(ga0, &As[0][frow][fcol]);
    stage_b128(ga1, &As[0][frow + 32][fcol]);
    stage_b128(gb0, &Bs[0][frow][fcol]);
    stage_b128(gb1, &Bs[0][frow + 32][fcol]);
    stage_wait();
    __syncthreads();

    int buf = 0;
    for (int k0 = 0; k0 < Nn; k0 += KB) {
        // prefetch next K-block into the other buffer (overlaps WMMA below)
        if (k0 + KB < Nn) {
            const int nk = k0 + KB;
            stage_b128(ga0 + nk, &As[buf ^ 1][frow][fcol]);
            stage_b128(ga1 + nk, &As[buf ^ 1][frow + 32][fcol]);
            stage_b128(gb0 + nk, &Bs[buf ^ 1][frow][fcol]);
            stage_b128(gb1 + nk, &Bs[buf ^ 1][frow + 32][fcol]);
        }

        #pragma unroll
        for (int kk = 0; kk < KB; kk += 4) {
            v2f a  = *(const v2f*)&As[buf][mA][kk + kh];
            v2f b0 = *(const v2f*)&Bs[buf][nB][kk + kh];
            v2f b1 = *(const v2f*)&Bs[buf][nB + 16][kk + kh];
            acc0 = __builtin_amdgcn_wmma_f32_16x16x4_f32(false, a, false, b0,
                                                         (short)0, acc0, false, false);
            acc1 = __builtin_amdgcn_wmma_f32_16x16x4_f32(false, a, false, b1,
                                                         (short)0, acc1, false, false);
        }

        stage_wait();
        __syncthreads();
        buf ^= 1;
    }

    // C/D layout: lane L -> N = L&15 ; VGPR i -> M = i + (L>>4)*8
    const int nOut  = lane & 15;
    const int mBase = (lane >> 4) * 8;
    #pragma unroll
    for (int i = 0; i < 8; ++i) {
        const int m = tileM + waveM + mBase + i;
        E[(size_t)m * Cc + tileN + waveN + nOut]      = acc0[i];
        E[(size_t)m * Cc + tileN + waveN + 16 + nOut] = acc1[i];
    }
}

// ---------------------------------------------------------------------------
// Kernel 2: in-place reverse-max softmax over rows of energy.
// softmax(rowmax - e) == exp(rowmin - e) / sum(exp(rowmin - e))
// One wave per 512-element row; 8 rows per 256-thread block.
// ---------------------------------------------------------------------------
__global__ __launch_bounds__(256)
void cam_softmax_kernel(float* __restrict__ energy) {
    const int row  = blockIdx.x * 8 + (threadIdx.x >> 5);   // 0 .. 16*512-1
    const int lane = threadIdx.x & 31;
    float* E = energy + (size_t)row * Cc;

    float v[16];
    float mn = 3.402823466e38f;
    #pragma unroll
    for (int i = 0; i < 16; ++i) {
        v[i] = E[lane + i * 32];
        mn = fminf(mn, v[i]);
    }
    #pragma unroll
    for (int off = 16; off > 0; off >>= 1)
        mn = fminf(mn, __shfl_xor(mn, off, 32));

    float s = 0.0f;
    #pragma unroll
    for (int i = 0; i < 16; ++i) {
        v[i] = __expf(mn - v[i]);
        s += v[i];
    }
    #pragma unroll
    for (int off = 16; off > 0; off >>= 1)
        s += __shfl_xor(s, off, 32);

    const float inv = 1.0f / s;
    #pragma unroll
    for (int i = 0; i < 16; ++i)
        E[lane + i * 32] = v[i] * inv;
}

// ---------------------------------------------------------------------------
// Kernel 3: out = gamma * (attention @ Q) + x
// M = C (512), K = C (512), N = 4096. 64x64 tile per WG, double-buffered LDS.
// ---------------------------------------------------------------------------
__global__ __launch_bounds__(256)
void cam_out_kernel(const float* __restrict__ x, const float* __restrict__ attn,
                    const float* __restrict__ gamma, float* __restrict__ out) {
    const int b     = blockIdx.z;
    const int tileM = blockIdx.y * 64;   // over C
    const int tileN = blockIdx.x * 64;   // over N
    const float* Q = x    + (size_t)b * Cc * Nn;
    const float* A = attn + (size_t)b * Cc * Cc;

    __shared__ __align__(16) float As[2][64][LP];       // attention tile [m][k]
    __shared__ __align__(16) float Bs[2][KB][64 + 4];   // Q tile [k][n] (stride 68 = 272B, 16B-aligned)

    const int tid   = threadIdx.x;
    const int lane  = tid & 31;
    const int wave  = tid >> 5;
    const int waveM = (wave & 3) * 16;
    const int waveN = (wave >> 2) * 32;

    const int frow  = tid >> 3;          // A fill: 0..31
    const int fcol  = (tid & 7) * 4;
    const int gkrow = tid >> 4;          // B fill: 0..15
    const int gncol = (tid & 15) * 4;

    const int mA = waveM + (lane & 15);
    const int nB = waveN + (lane & 15);
    const int kh = (lane >> 4) * 2;

    v8f acc0 = {}; v8f acc1 = {};

    const float* gaA0 = A + (size_t)(tileM + frow)      * Cc + fcol;
    const float* gaA1 = A + (size_t)(tileM + frow + 32) * Cc + fcol;
    const float* gqB0 = Q + (size_t)(gkrow)      * Nn + tileN + gncol;
    const float* gqB1 = Q + (size_t)(gkrow + 16) * Nn + tileN + gncol;

    stage_b128(gaA0, &As[0][frow][fcol]);
    stage_b128(gaA1, &As[0][frow + 32][fcol]);
    stage_b128(gqB0, &Bs[0][gkrow][gncol]);
    stage_b128(gqB1, &Bs[0][gkrow + 16][gncol]);
    stage_wait();
    __syncthreads();

    int buf = 0;
    for (int k0 = 0; k0 < Cc; k0 += KB) {
        if (k0 + KB < Cc) {
            const int nk = k0 + KB;
            stage_b128(gaA0 + nk, &As[buf ^ 1][frow][fcol]);
            stage_b128(gaA1 + nk, &As[buf ^ 1][frow + 32][fcol]);
            stage_b128(gqB0 + (size_t)nk * Nn, &Bs[buf ^ 1][gkrow][gncol]);
            stage_b128(gqB1 + (size_t)nk * Nn, &Bs[buf ^ 1][gkrow + 16][gncol]);
        }

        #pragma unroll
        for (int kk = 0; kk < KB; kk += 4) {
            v2f a = *(const v2f*)&As[buf][mA][kk + kh];
            v2f b0, b1;
            b0.x = Bs[buf][kk + kh][nB];       b0.y = Bs[buf][kk + kh + 1][nB];
            b1.x = Bs[buf][kk + kh][nB + 16];  b1.y = Bs[buf][kk + kh + 1][nB + 16];
            acc0 = __builtin_amdgcn_wmma_f32_16x16x4_f32(false, a, false, b0,
                                                         (short)0, acc0, false, false);
            acc1 = __builtin_amdgcn_wmma_f32_16x16x4_f32(false, a, false, b1,
                                                         (short)0, acc1, false, false);
        }

        stage_wait();
        __syncthreads();
        buf ^= 1;
    }

    const float g = gamma[0];
    const int nOut  = lane & 15;
    const int mBase = (lane >> 4) * 8;
    #pragma unroll
    for (int i = 0; i < 8; ++i) {
        const int m = tileM + waveM + mBase + i;
        const size_t base = ((size_t)b * Cc + m) * Nn + tileN + waveN + nOut;
        out[base]      = g * acc0[i] + x[base];
        out[base + 16] = g * acc1[i] + x[base + 16];
    }
}

// ---------------------------------------------------------------------------
extern "C" void kernel_launch(void* const* d_in, const int* in_sizes, int n_in,
                              void* d_out, int out_size, void* d_ws, size_t ws_size,
                              hipStream_t stream) {
    (void)in_sizes; (void)n_in; (void)out_size; (void)ws_size;
    const float* x     = (const float*)d_in[0];
    const float* gamma = (const float*)d_in[1];
    float*       out   = (float*)d_out;
    float*       energy = (float*)d_ws;   // 16*512*512 floats = 16.8 MB scratch

    cam_energy_kernel<<<dim3(8, 8, 16), 256, 0, stream>>>(x, energy);
    cam_softmax_kernel<<<dim3((16 * Cc) / 8), 256, 0, stream>>>(energy);
    cam_out_kernel<<<dim3(Nn / 64, Cc / 64, 16), 256, 0, stream>>>(x, energy, gamma, out);
}